// VectorQuantizer_77721728189142
// MI455X (gfx1250) — compile-verified
//
#include <hip/hip_runtime.h>

typedef __attribute__((ext_vector_type(2))) float v2f;
typedef __attribute__((ext_vector_type(4))) float v4f;
typedef __attribute__((ext_vector_type(8))) float v8f;

#define HW_SZ    4096       // 64*64
#define C_DIM    64
#define K_CODES  1024
#define N_PIX    65536      // 16 * 4096
#define TOT_ELEMS 4194304   // N_PIX * C_DIM

// ---------------- Kernel A: codebook row norms ----------------
__global__ void vq_enorm(const float* __restrict__ embed, float* __restrict__ enorm) {
    int k = blockIdx.x * blockDim.x + threadIdx.x;
    if (k >= K_CODES) return;
    const v4f* row = (const v4f*)(embed + (size_t)k * C_DIM);
    float s = 0.f;
#pragma unroll
    for (int i = 0; i < 16; ++i) {
        v4f v = row[i];
        s += v.x * v.x + v.y * v.y + v.z * v.z + v.w * v.w;
    }
    enorm[k] = s;
}

// ---------------- Kernel B: [B,C,HW] -> z_flat [B*HW, C] ----------------
__global__ void vq_transpose(const float* __restrict__ z, float* __restrict__ zf) {
    __shared__ float tile[32][33];
    const int b   = blockIdx.z;
    const int hw0 = blockIdx.x * 32;
    const int c0  = blockIdx.y * 32;
    const float* src = z + (size_t)b * C_DIM * HW_SZ;
#pragma unroll
    for (int j = 0; j < 32; j += 8)
        tile[threadIdx.y + j][threadIdx.x] =
            src[(size_t)(c0 + threadIdx.y + j) * HW_SZ + hw0 + threadIdx.x];
    __syncthreads();
    float* dst = zf + ((size_t)b * HW_SZ + hw0) * C_DIM + c0;
#pragma unroll
    for (int j = 0; j < 32; j += 8)
        dst[(size_t)(threadIdx.y + j) * C_DIM + threadIdx.x] =
            tile[threadIdx.x][threadIdx.y + j];
}

// ---------------- Kernel C: fp32 WMMA distance GEMM + argmin ----------------
// One wave handles an M=32 tile (two 16-row WMMA tiles) against all 1024 codes;
// each B fragment feeds two WMMAs (16 FLOP/byte from cache). The whole 16-fragment
// B tile is staged in registers so loads clause together and overlap the WMMA drain.
// A (16x4 f32) layout: lane m = lane&15, VGPR pair holds K = (lane>>4)*2, +1.
// B (4x16 f32) layout: lane n = lane&15 (code col), same K-pair selection.
// C/D (16x16 f32): VGPR r, lanes 0-15 -> M=r; lanes 16-31 -> M=8+r; N=lane&15.
__global__ void __launch_bounds__(128, 1)
vq_argmin(const float* __restrict__ zf, const float* __restrict__ embed,
          const float* __restrict__ enorm, int* __restrict__ idxOut) {
    const int lane = threadIdx.x & 31;
    const int wave = threadIdx.x >> 5;
    const int tile = blockIdx.x * 4 + wave;
    const int m0   = tile * 32;
    const int half = lane >> 4;   // 0 or 1
    const int lm   = lane & 15;
    const int kb   = half * 2;    // K offset within each group of 4

    // Preload two A tiles: 16 float2 per lane each (K = 0..63 in this lane's slots)
    v2f areg0[16], areg1[16];
    const float* arow0 = zf + (size_t)(m0 + lm) * C_DIM + kb;
    const float* arow1 = zf + (size_t)(m0 + 16 + lm) * C_DIM + kb;
#pragma unroll
    for (int t = 0; t < 16; ++t) {
        areg0[t] = *(const v2f*)(arow0 + 4 * t);
        areg1[t] = *(const v2f*)(arow1 + 4 * t);
    }

    float bestv0[8], bestv1[8];
    int   besti0[8], besti1[8];
#pragma unroll
    for (int r = 0; r < 8; ++r) {
        bestv0[r] = __builtin_inff(); besti0[r] = 0;
        bestv1[r] = __builtin_inff(); besti1[r] = 0;
    }

    for (int nt = 0; nt < 64; ++nt) {
        const int code = nt * 16 + lm;             // this lane's code column
        const float* brow = embed + (size_t)code * C_DIM + kb;
        const float en = enorm[code];

        // Stage full B tile in registers: loads can issue as one clause and
        // retire while the previous iteration's WMMAs drain.
        v2f bv[16];
#pragma unroll
        for (int t = 0; t < 16; ++t)
            bv[t] = *(const v2f*)(brow + 4 * t);

        v8f acc0 = {};
        v8f acc1 = {};
#pragma unroll
        for (int t = 0; t < 16; ++t) {
            acc0 = __builtin_amdgcn_wmma_f32_16x16x4_f32(
                false, areg0[t], false, bv[t], (short)0, acc0, false, false);
            acc1 = __builtin_amdgcn_wmma_f32_16x16x4_f32(
                false, areg1[t], false, bv[t], (short)0, acc1, false, false);
        }
#pragma unroll
        for (int r = 0; r < 8; ++r) {
            float d0 = en - 2.0f * acc0[r];        // znorm omitted: constant per row
            float d1 = en - 2.0f * acc1[r];
            if (d0 < bestv0[r]) { bestv0[r] = d0; besti0[r] = code; }  // increasing idx -> first-min
            if (d1 < bestv1[r]) { bestv1[r] = d1; besti1[r] = code; }
        }
    }

    // Min+argmin across the 16 lanes holding each row's columns (xor stays in half)
#pragma unroll
    for (int off = 1; off < 16; off <<= 1) {
#pragma unroll
        for (int r = 0; r < 8; ++r) {
            float ov0 = __shfl_xor(bestv0[r], off, 32);
            int   oi0 = __shfl_xor(besti0[r], off, 32);
            if (ov0 < bestv0[r] || (ov0 == bestv0[r] && oi0 < besti0[r])) {
                bestv0[r] = ov0; besti0[r] = oi0;
            }
            float ov1 = __shfl_xor(bestv1[r], off, 32);
            int   oi1 = __shfl_xor(besti1[r], off, 32);
            if (ov1 < bestv1[r] || (ov1 == bestv1[r] && oi1 < besti1[r])) {
                bestv1[r] = ov1; besti1[r] = oi1;
            }
        }
    }
    if (lm == 0) {
#pragma unroll
        for (int r = 0; r < 8; ++r) {
            idxOut[m0 + half * 8 + r]      = besti0[r];  // lane0: rows 0..7, lane16: rows 8..15
            idxOut[m0 + 16 + half * 8 + r] = besti1[r];  // second 16-row tile
        }
    }
}

// ---------------- Kernel D: gather z_q, write straight-through out, loss partials ----------------
__global__ void vq_gather(const float* __restrict__ zf, const float* __restrict__ embed,
                          const int* __restrict__ idx, float* __restrict__ out,
                          float* __restrict__ partials) {
    __shared__ float red[256];
    const int n  = blockIdx.x * 256 + threadIdx.x;
    const int b  = n >> 12;      // / 4096
    const int hw = n & 4095;
    const int code = idx[n];
    const float* erow = embed + (size_t)code * C_DIM;
    const float* zrow = zf + (size_t)n * C_DIM;
    float* obase = out + (size_t)b * C_DIM * HW_SZ + hw;

    float acc = 0.f;
#pragma unroll
    for (int c = 0; c < C_DIM; ++c) {
        float e  = erow[c];
        float zv = zrow[c];
        float dq = e - zv;
        obase[(size_t)c * HW_SZ] = zv + dq;   // z + (z_q - z), matching reference fp math
        acc += dq * dq;
    }
    red[threadIdx.x] = acc;
    __syncthreads();
    for (int s = 128; s > 0; s >>= 1) {
        if (threadIdx.x < s) red[threadIdx.x] += red[threadIdx.x + s];
        __syncthreads();
    }
    if (threadIdx.x == 0) partials[blockIdx.x] = red[0];
}

// ---------------- Kernel E: deterministic final loss reduction ----------------
__global__ void vq_loss(const float* __restrict__ partials, float* __restrict__ lossOut) {
    __shared__ float red[256];
    red[threadIdx.x] = partials[threadIdx.x];
    __syncthreads();
    for (int s = 128; s > 0; s >>= 1) {
        if (threadIdx.x < s) red[threadIdx.x] += red[threadIdx.x + s];
        __syncthreads();
    }
    if (threadIdx.x == 0)
        lossOut[0] = 1.25f * red[0] / (float)TOT_ELEMS;
}

extern "C" void kernel_launch(void* const* d_in, const int* in_sizes, int n_in,
                              void* d_out, int out_size, void* d_ws, size_t ws_size,
                              hipStream_t stream) {
    const float* z     = (const float*)d_in[0];   // [16,64,64,64] fp32
    const float* embed = (const float*)d_in[1];   // [1024,64] fp32
    float* out = (float*)d_out;                   // [16,64,64,64] + 1 loss scalar

    char* ws = (char*)d_ws;
    float* zf       = (float*)ws;                                         // 16 MB
    float* enorm    = (float*)(ws + (size_t)TOT_ELEMS * 4);               // 4 KB
    int*   idx      = (int*)  (ws + (size_t)TOT_ELEMS * 4 + 4096);        // 256 KB
    float* partials = (float*)(ws + (size_t)TOT_ELEMS * 4 + 4096 + (size_t)N_PIX * 4); // 1 KB

    vq_enorm<<<K_CODES / 256, 256, 0, stream>>>(embed, enorm);

    dim3 tb(32, 8);
    dim3 tg(HW_SZ / 32, C_DIM / 32, 16);
    vq_transpose<<<tg, tb, 0, stream>>>(z, zf);

    vq_argmin<<<N_PIX / 32 / 4, 128, 0, stream>>>(zf, embed, enorm, idx);

    vq_gather<<<N_PIX / 256, 256, 0, stream>>>(zf, embed, idx, out, partials);

    vq_loss<<<1, 256, 0, stream>>>(partials, out + TOT_ELEMS);
}